// HandPoseLegitimizeLayer_34239479284009
// MI455X (gfx1250) — compile-verified
//
#include <hip/hip_runtime.h>
#include <hip/hip_bf16.h>
#include <stdint.h>

#define EPSF 1e-6f
#define TPB 256
#define JFLOATS 63

typedef __attribute__((ext_vector_type(4))) unsigned int v4u;
typedef __attribute__((ext_vector_type(8))) int v8i;
typedef __attribute__((ext_vector_type(4))) int v4i;

struct V3 { float x, y, z; };

__device__ __forceinline__ V3 vadd(V3 a, V3 b){ return {a.x+b.x, a.y+b.y, a.z+b.z}; }
__device__ __forceinline__ V3 vsub(V3 a, V3 b){ return {a.x-b.x, a.y-b.y, a.z-b.z}; }
__device__ __forceinline__ V3 vscale(V3 a, float s){ return {a.x*s, a.y*s, a.z*s}; }
__device__ __forceinline__ float vdot(V3 a, V3 b){ return a.x*b.x + a.y*b.y + a.z*b.z; }
__device__ __forceinline__ V3 vcross(V3 a, V3 b){
  return {a.y*b.z - a.z*b.y, a.z*b.x - a.x*b.z, a.x*b.y - a.y*b.x};
}
__device__ __forceinline__ V3 vunit(V3 a){
  float n = sqrtf(vdot(a,a)) + EPSF;
  return vscale(a, 1.0f/n);
}
__device__ __forceinline__ float clipc(float d){
  return fminf(fmaxf(d, -1.0f + EPSF), 1.0f - EPSF);
}

__device__ __forceinline__ V3 matv(const float R[9], V3 v){
  return { R[0]*v.x + R[1]*v.y + R[2]*v.z,
           R[3]*v.x + R[4]*v.y + R[5]*v.z,
           R[6]*v.x + R[7]*v.y + R[8]*v.z };
}

// Rodrigues rotation matrix, axis normalized by (|axis|+EPS), matching reference.
__device__ __forceinline__ void rodrigues1(V3 axis, float t, float R[9]){
  float n = sqrtf(vdot(axis,axis)) + EPSF;
  float inv = 1.0f/n;
  float x = axis.x*inv, y = axis.y*inv, z = axis.z*inv;
  float c = cosf(t), s = sinf(t), C = 1.0f - c;
  R[0]=c+x*x*C;   R[1]=x*y*C-z*s; R[2]=x*z*C+y*s;
  R[3]=x*y*C+z*s; R[4]=c+y*y*C;   R[5]=y*z*C-x*s;
  R[6]=x*z*C-y*s; R[7]=y*z*C+x*s; R[8]=c+z*z*C;
}

// Build R(+t) and R(-t) with one sincos.
__device__ __forceinline__ void rodrigues2(V3 axis, float t, float Rp[9], float Rm[9]){
  float n = sqrtf(vdot(axis,axis)) + EPSF;
  float inv = 1.0f/n;
  float x = axis.x*inv, y = axis.y*inv, z = axis.z*inv;
  float c = cosf(t), s = sinf(t), C = 1.0f - c;
  float xx=x*x*C, yy=y*y*C, zz=z*z*C;
  float xy=x*y*C, xz=x*z*C, yz=y*z*C;
  float xs=x*s, ys=y*s, zs=z*s;
  Rp[0]=c+xx; Rp[1]=xy-zs; Rp[2]=xz+ys;
  Rp[3]=xy+zs; Rp[4]=c+yy; Rp[5]=yz-xs;
  Rp[6]=xz-ys; Rp[7]=yz+xs; Rp[8]=c+zz;
  Rm[0]=c+xx; Rm[1]=xy+zs; Rm[2]=xz-ys;
  Rm[3]=xy-zs; Rm[4]=c+yy; Rm[5]=yz+xs;
  Rm[6]=xz+ys; Rm[7]=yz-xs; Rm[8]=c+zz;
}

__device__ __forceinline__ void getrot(float angleP, float angle, float flexRatio,
                                       V3 axis, V3 sv, V3 ev, float R[9]){
  const float TH = 3.1415926f/180.0f*120.0f;
  angle = (angle > TH) ? (3.1415926f - angle) : angle;
  float dif = fmaxf(angle - angleP, 0.0f) * flexRatio;
  float R0[9], R1[9];
  rodrigues2(axis, dif, R0, R1);
  V3 v0 = vunit(matv(R0, sv));
  V3 v1 = vunit(matv(R1, sv));
  bool m0 = fabsf(vdot(v0, ev)) > fabsf(vdot(v1, ev));
  #pragma unroll
  for (int k=0;k<9;++k) R[k] = m0 ? R0[k] : R1[k];
}

// Smallest-eigenvalue eigenvector of symmetric 3x3 (trigonometric closed form).
__device__ __forceinline__ V3 smallest_evec(float a00,float a01,float a02,
                                            float a11,float a12,float a22){
  float p1 = a01*a01 + a02*a02 + a12*a12;
  float q  = (a00+a11+a22)*(1.0f/3.0f);
  float b00=a00-q, b11=a11-q, b22=a22-q;
  float p2 = b00*b00 + b11*b11 + b22*b22 + 2.0f*p1;
  float p  = sqrtf(p2*(1.0f/6.0f));
  float lam = q;
  if (p > 1e-12f) {
    float inv = 1.0f/p;
    float c00=b00*inv, c01=a01*inv, c02=a02*inv;
    float c11=b11*inv, c12=a12*inv, c22=b22*inv;
    float detB = c00*(c11*c22 - c12*c12)
               - c01*(c01*c22 - c12*c02)
               + c02*(c01*c12 - c11*c02);
    float r = fminf(fmaxf(0.5f*detB, -1.0f), 1.0f);
    float phi = acosf(r)*(1.0f/3.0f);
    lam = q + 2.0f*p*cosf(phi + 2.0943951023931953f); // +2pi/3 -> smallest
  }
  V3 r0 = {a00-lam, a01, a02};
  V3 r1 = {a01, a11-lam, a12};
  V3 r2 = {a02, a12, a22-lam};
  V3 c01v = vcross(r0,r1), c02v = vcross(r0,r2), c12v = vcross(r1,r2);
  float n01 = vdot(c01v,c01v), n02 = vdot(c02v,c02v), n12 = vdot(c12v,c12v);
  V3 best = c01v; float bn = n01;
  if (n02 > bn){ best = c02v; bn = n02; }
  if (n12 > bn){ best = c12v; bn = n12; }
  if (bn < 1e-24f) return {0.0f, 0.0f, 1.0f};
  return vscale(best, 1.0f/sqrtf(bn));
}

__device__ __forceinline__ V3 palmNorm(const V3 j[21], int i){
  const int MCPC[5] = {1,4,10,7,13};
  V3 w = j[0];
  V3 a, b;
  if (i < 4) { a = vsub(j[MCPC[i]], w); b = vsub(j[MCPC[(i+1)&3]], w); }
  else       { a = vsub(j[13], w);      b = vsub(j[1],  w); }
  return vunit(vcross(a, b));
}

__device__ __forceinline__ void planarize(V3 j[21]){
  const int IDX[5][4] = {{1,2,3,17},{4,5,6,18},{10,11,12,19},{7,8,9,20},{13,14,15,16}};
  #pragma unroll
  for (int f=0; f<5; ++f){
    V3 p0=j[IDX[f][0]], p1=j[IDX[f][1]], p2=j[IDX[f][2]], p3=j[IDX[f][3]];
    V3 c = vscale(vadd(vadd(p0,p1), vadd(p2,p3)), 0.25f);
    float a00=0,a01=0,a02=0,a11=0,a12=0,a22=0;
    V3 P[4] = {p0,p1,p2,p3};
    #pragma unroll
    for (int k=0;k<4;++k){
      V3 d = vsub(P[k], c);
      a00 += d.x*d.x; a01 += d.x*d.y; a02 += d.x*d.z;
      a11 += d.y*d.y; a12 += d.y*d.z; a22 += d.z*d.z;
    }
    V3 nrm = smallest_evec(a00,a01,a02,a11,a12,a22);
    float vd = -vdot(nrm, c);
    #pragma unroll
    for (int k=0;k<4;++k){
      float dist = vdot(P[k], nrm) + vd;
      j[IDX[f][k]] = vsub(P[k], vscale(nrm, dist));
    }
  }
}

__device__ __forceinline__ void abduction(V3 j[21]){
  const int MCPC[5]={1,4,10,7,13}, PIPC[5]={2,5,11,8,14};
  const int ABCH[5][3]={{2,3,17},{5,6,18},{11,12,19},{8,9,20},{14,15,16}};
  const float RECT[5]={0.189f,0.1331f,-0.1491f,0.0347f,0.0f};
  const float angleP = 0.3490658503988659f; // pi/9
  #pragma unroll
  for (int i=0;i<5;++i){
    V3 palm = palmNorm(j, i);
    V3 mcp = j[MCPC[i]], pip = j[PIPC[i]], wrist = j[0];
    float vd = -vdot(mcp, palm);
    float dist = vdot(pip, palm) + vd;
    V3 projpip = vsub(pip, vscale(palm, dist));
    V3 mp = vsub(mcp, pip);
    float dis = sqrtf(vdot(mp, mp));
    V3 pm = vsub(projpip, mcp);
    float flex = sqrtf(vdot(pm, pm)) / (dis + EPSF);
    flex = (flex < 0.3f) ? 0.0f : flex;
    V3 wristmcp = vunit(vsub(mcp, wrist));
    V3 mcpproj  = vunit(pm);
    V3 mcppip   = vunit(vsub(pip, mcp));
    bool overflex = acosf(clipc(vdot(wristmcp, mcppip))) > 1.57f; // 3.14/2
    float Rr[9];
    rodrigues1(palm, RECT[i], Rr);
    V3 rect = matv(Rr, wristmcp);
    float ang = acosf(clipc(vdot(rect, mcpproj)));
    ang = overflex ? 0.0f : ang;
    float R[9];
    getrot(angleP, ang, flex, palm, mcpproj, wristmcp, R);
    #pragma unroll
    for (int k=0;k<3;++k){
      int ch = ABCH[i][k];
      V3 v = vsub(j[ch], mcp);
      j[ch] = vadd(matv(R, v), mcp);
    }
  }
}

__device__ __forceinline__ void planeRotation(V3 j[21]){
  const int MCPC[5]={1,4,10,7,13}, PIPC[5]={2,5,11,8,14}, DIPC[5]={3,6,12,9,15};
  const int RESTC[4][2]={{3,17},{6,18},{12,19},{9,20}};
  const float angleN = 0.3490658503988659f; // pi/9
  #pragma unroll
  for (int i=0;i<4;++i){
    V3 mcppip = vunit(vsub(j[PIPC[i]], j[MCPC[i]]));
    V3 pipdip = vunit(vsub(j[DIPC[i]], j[PIPC[i]]));
    bool maskline = fabsf(vdot(mcppip, pipdip)) > 0.95f;
    V3 cdir = vunit(vcross(mcppip, pipdip));
    V3 palm = palmNorm(j, i);
    V3 wm = vunit(vsub(j[MCPC[i]], j[0]));
    V3 stdd = vunit(vcross(wm, palm));
    float ang = acosf(clipc(vdot(cdir, stdd)));
    ang = maskline ? 0.0f : ang;
    float R[9];
    getrot(angleN, ang, 1.0f, mcppip, cdir, stdd, R);
    V3 pivot = j[PIPC[i]];
    #pragma unroll
    for (int k=0;k<2;++k){
      int ch = RESTC[i][k];
      V3 v = vsub(j[ch], pivot);
      j[ch] = vadd(matv(R, v), pivot);
    }
  }
}

// Build TDM descriptor groups for a 1-D tile of nFloats f32 (ISA cdna5 §8).
__device__ __forceinline__ void build_tdm(unsigned ldsAddr, unsigned long long gaddr,
                                          unsigned nFloats, v4u* g0, v8i* g1){
  (*g0)[0] = 1u;                                            // count=1, is_restore=0
  (*g0)[1] = ldsAddr;                                       // lds_addr
  (*g0)[2] = (unsigned)(gaddr & 0xffffffffull);             // global_addr lo
  (*g0)[3] = (unsigned)((gaddr >> 32) & 0x1ffffffull)       // global_addr hi (57-bit)
           | (2u << 30);                                    // type=2 ("image")
  unsigned h0 = (2u << 16);                                 // data_size=2 -> 4 bytes
  unsigned h1 = (nFloats & 0xffffu) << 16;                  // tensor_dim0[15:0] @ bits 63:48
  unsigned h2 = ((nFloats >> 16) & 0xffffu) | (1u << 16);   // td0 hi | tensor_dim1=1
  unsigned h3 = (nFloats & 0xffffu) << 16;                  // tile_dim0 @ bits 127:112
  unsigned h4 = 1u;                                         // tile_dim1=1, tile_dim2=0
  unsigned h5 = nFloats;                                    // tensor_dim0_stride lo32
  (*g1)[0]=(int)h0; (*g1)[1]=(int)h1; (*g1)[2]=(int)h2; (*g1)[3]=(int)h3;
  (*g1)[4]=(int)h4; (*g1)[5]=(int)h5; (*g1)[6]=0; (*g1)[7]=0;
}

__global__ __launch_bounds__(TPB)
void handpose_kernel(const float* __restrict__ in, float* __restrict__ out, int n){
  __shared__ float lds[TPB * JFLOATS];   // 64512 B, well under 320 KB/WGP
  const int tid = threadIdx.x;
  const int blockBase = (int)blockIdx.x * TPB;
  int blockElems = n - blockBase;
  blockElems = (blockElems > TPB) ? TPB : blockElems;
  const unsigned nFloats = (unsigned)(blockElems * JFLOATS);
  const unsigned ldsAddr = (unsigned)(unsigned long long)(uintptr_t)&lds[0];

  // ---- TDM: DMA this block's tile global -> LDS (wave 0 only; branch is wave-uniform)
  if (tid < 32) {
    v4u g0; v8i g1; v4i gz4 = {0,0,0,0}; v8i gz8 = {0,0,0,0,0,0,0,0};
    build_tdm(ldsAddr,
              (unsigned long long)(uintptr_t)(in + (size_t)blockBase * JFLOATS),
              nFloats, &g0, &g1);
    __builtin_amdgcn_tensor_load_to_lds(g0, g1, gz4, gz4, gz8, 0);
    __builtin_amdgcn_s_wait_tensorcnt(0);
  }
  __syncthreads();
  asm volatile("" ::: "memory");   // LDS now holds TDM-written data

  if (tid < blockElems) {
    V3 j[21];
    float* my = &lds[tid * JFLOATS];   // stride 63 words: conflict-free on 64 banks
    #pragma unroll
    for (int k=0;k<21;++k) j[k] = (V3){ my[3*k+0], my[3*k+1], my[3*k+2] };

    planarize(j);
    abduction(j);
    planeRotation(j);
    planarize(j);

    #pragma unroll
    for (int k=0;k<21;++k){ my[3*k+0]=j[k].x; my[3*k+1]=j[k].y; my[3*k+2]=j[k].z; }
  }
  asm volatile("" ::: "memory");
  __syncthreads();

  // ---- TDM: DMA results LDS -> global
  if (tid < 32) {
    v4u g0; v8i g1; v4i gz4 = {0,0,0,0}; v8i gz8 = {0,0,0,0,0,0,0,0};
    build_tdm(ldsAddr,
              (unsigned long long)(uintptr_t)(out + (size_t)blockBase * JFLOATS),
              nFloats, &g0, &g1);
    __builtin_amdgcn_tensor_store_from_lds(g0, g1, gz4, gz4, gz8, 0);
    __builtin_amdgcn_s_wait_tensorcnt(0);
  }
}

extern "C" void kernel_launch(void* const* d_in, const int* in_sizes, int n_in,
                              void* d_out, int out_size, void* d_ws, size_t ws_size,
                              hipStream_t stream) {
  const float* joints = (const float*)d_in[0];
  float* out = (float*)d_out;
  const int n = in_sizes[0] / JFLOATS;    // 500000 elements of 21x3 f32
  const int grid = (n + TPB - 1) / TPB;
  handpose_kernel<<<grid, TPB, 0, stream>>>(joints, out, n);
}